// MultiHeadAttention_53403623358601
// MI455X (gfx1250) — compile-verified
//
#include <hip/hip_runtime.h>

typedef __attribute__((ext_vector_type(2))) float v2f;
typedef __attribute__((ext_vector_type(8))) float v8f;

#define D_MODEL 128
#define NUM_HEADS 8
#define DEPTH 16
#define SEQ 2048
#define BATCH 4

__device__ __forceinline__ v8f wmma_f32(v2f a, v2f b, v8f c) {
  // D = A(16x4,f32) * B(4x16,f32) + C(16x16,f32)
  return __builtin_amdgcn_wmma_f32_16x16x4_f32(false, a, false, b, (short)0, c,
                                               false, false);
}

// ---------------------------------------------------------------------------
// X[rows x 128] @ W[128 x 128] + bias, scattered to head layout [B,H,S,16].
// One wave per 16x16 output tile; 8 waves cover the 128 output columns.
// ---------------------------------------------------------------------------
__global__ __launch_bounds__(256) void proj_to_heads(
    const float* __restrict__ x, const float* __restrict__ W,
    const float* __restrict__ bias, float* __restrict__ outHead) {
  const int lane = threadIdx.x & 31;
  const int wave = threadIdx.x >> 5;
  const int half = lane >> 4;   // 0: K={0,1}, 1: K={2,3}
  const int lm = lane & 15;
  const int row0 = blockIdx.x * 16;
  const int col0 = wave * 16;   // head h == wave (depth 16 aligned tiles)

  v8f c = {};
  const float* xr = x + (size_t)(row0 + lm) * D_MODEL;
  for (int kk = 0; kk < D_MODEL; kk += 4) {
    const int kd = kk + half * 2;
    v2f a = {xr[kd], xr[kd + 1]};
    v2f b = {W[(size_t)kd * D_MODEL + col0 + lm],
             W[(size_t)(kd + 1) * D_MODEL + col0 + lm]};
    c = wmma_f32(a, b, c);
  }
  const float bv = bias[col0 + lm];
#pragma unroll
  for (int j = 0; j < 8; ++j) {
    const int m = j + half * 8;
    const int r = row0 + m;           // global row in [0, B*S)
    const int bidx = r / SEQ;
    const int s = r % SEQ;
    outHead[(((size_t)bidx * NUM_HEADS + wave) * SEQ + s) * DEPTH + lm] =
        c[j] + bv;
  }
}

// ---------------------------------------------------------------------------
// dense: ctx_flat[8192 x 128] @ dense_w[128 x 128] + dense_b -> out row-major
// ---------------------------------------------------------------------------
__global__ __launch_bounds__(256) void dense_out(
    const float* __restrict__ x, const float* __restrict__ W,
    const float* __restrict__ bias, float* __restrict__ out) {
  const int lane = threadIdx.x & 31;
  const int wave = threadIdx.x >> 5;
  const int half = lane >> 4;
  const int lm = lane & 15;
  const int row0 = blockIdx.x * 16;
  const int col0 = wave * 16;

  v8f c = {};
  const float* xr = x + (size_t)(row0 + lm) * D_MODEL;
  for (int kk = 0; kk < D_MODEL; kk += 4) {
    const int kd = kk + half * 2;
    v2f a = {xr[kd], xr[kd + 1]};
    v2f b = {W[(size_t)kd * D_MODEL + col0 + lm],
             W[(size_t)(kd + 1) * D_MODEL + col0 + lm]};
    c = wmma_f32(a, b, c);
  }
  const float bv = bias[col0 + lm];
#pragma unroll
  for (int j = 0; j < 8; ++j) {
    const int m = j + half * 8;
    out[(size_t)(row0 + m) * D_MODEL + col0 + lm] = c[j] + bv;
  }
}

// ---------------------------------------------------------------------------
// Attention: one block = (b, h, 16-row q tile). Full 16x2048 logits strip in
// LDS (128 KB of the 320 KB WGP LDS). Softmax in LDS, one coalesced attn
// write, then ctx = P @ V via WMMA with a cross-wave LDS reduction.
// ---------------------------------------------------------------------------
__global__ __launch_bounds__(256) void attention(
    const float* __restrict__ qh, const float* __restrict__ kh,
    const float* __restrict__ vh, const float* __restrict__ mask,
    float* __restrict__ attn, float* __restrict__ ctx) {
  extern __shared__ float lds[];
  float* P = lds;                // 16 * 2048 logits/probs
  float* red = lds + 16 * SEQ;   // 256 reduction scratch
  float* ctxp = red + 256;       // 8 waves * 256 ctx partials

  const int tid = threadIdx.x;
  const int lane = tid & 31;
  const int wave = tid >> 5;
  const int half = lane >> 4;
  const int lm = lane & 15;

  const int bid = blockIdx.x;
  const int qt = bid & 127;         // 128 q-tiles per head
  const int h = (bid >> 7) & 7;
  const int b = bid >> 10;
  const int q0 = qt * 16;

  const size_t headOff = ((size_t)b * NUM_HEADS + h) * SEQ * DEPTH;
  const float* qb = qh + headOff;
  const float* kb = kh + headOff;
  const float* vb = vh + headOff;
  const float* mb = mask + (size_t)b * SEQ;

  // Q tile A-fragments: K dim = depth = 16 -> 4 WMMA steps
  v2f aq[4];
#pragma unroll
  for (int t = 0; t < 4; ++t) {
    const int kd = t * 4 + half * 2;
    const float* qr = qb + (size_t)(q0 + lm) * DEPTH + kd;
    aq[t] = {qr[0], qr[1]};
  }

  // logits: each wave owns keys [wave*256, wave*256+256)
  for (int ck = 0; ck < 16; ++ck) {
    const int kcol0 = wave * 256 + ck * 16;
    v8f c = {};
#pragma unroll
    for (int t = 0; t < 4; ++t) {
      const int kd = t * 4 + half * 2;
      const float* kr = kb + (size_t)(kcol0 + lm) * DEPTH + kd;
      v2f bf = {kr[0], kr[1]};
      c = wmma_f32(aq[t], bf, c);
    }
#pragma unroll
    for (int j = 0; j < 8; ++j) {
      const int row = j + half * 8;
      const int col = kcol0 + lm;
      P[row * SEQ + col] = c[j] * 0.25f + mb[col] * -1e9f;  // 1/sqrt(16)=0.25
    }
  }
  __syncthreads();

  // softmax: 16 threads per row
  const int srow = tid >> 4;
  const int sub = tid & 15;
  float mmax = -3.0e38f;
  for (int col = sub; col < SEQ; col += 16)
    mmax = fmaxf(mmax, P[srow * SEQ + col]);
  red[srow * 16 + sub] = mmax;
  __syncthreads();
  float rowmax = -3.0e38f;
#pragma unroll
  for (int i = 0; i < 16; ++i) rowmax = fmaxf(rowmax, red[srow * 16 + i]);
  __syncthreads();
  float s = 0.0f;
  for (int col = sub; col < SEQ; col += 16) {
    const float e = __expf(P[srow * SEQ + col] - rowmax);
    P[srow * SEQ + col] = e;
    s += e;
  }
  red[srow * 16 + sub] = s;
  __syncthreads();
  float rowsum = 0.0f;
#pragma unroll
  for (int i = 0; i < 16; ++i) rowsum += red[srow * 16 + i];
  const float inv = 1.0f / rowsum;
  for (int col = sub; col < SEQ; col += 16) P[srow * SEQ + col] *= inv;
  __syncthreads();

  // single coalesced write of this 16x2048 attn strip (the HBM-bound part)
  {
    float* ab = attn + (((size_t)b * NUM_HEADS + h) * SEQ + q0) * SEQ;
    for (int idx = tid; idx < 16 * SEQ; idx += 256) ab[idx] = P[idx];
  }

  // ctx = P(16x2048) @ V(2048x16): each wave accumulates its key slice
  v8f c2 = {};
  for (int t = 0; t < 64; ++t) {
    const int kg = wave * 256 + t * 4 + half * 2;
    v2f a = {P[lm * SEQ + kg], P[lm * SEQ + kg + 1]};
    const float* vr = vb + (size_t)kg * DEPTH + lm;
    v2f bf = {vr[0], vr[DEPTH]};
    c2 = wmma_f32(a, bf, c2);
  }
#pragma unroll
  for (int j = 0; j < 8; ++j) ctxp[wave * 256 + j * 32 + lane] = c2[j];
  __syncthreads();
  {
    float tot = 0.0f;
#pragma unroll
    for (int w = 0; w < 8; ++w) tot += ctxp[w * 256 + tid];
    const int j = tid >> 5;
    const int l = tid & 31;
    const int h2 = l >> 4;
    const int d = l & 15;
    const int row = j + h2 * 8;
    ctx[headOff + (size_t)(q0 + row) * DEPTH + d] = tot;
  }
}

// ---------------------------------------------------------------------------
extern "C" void kernel_launch(void* const* d_in, const int* in_sizes, int n_in,
                              void* d_out, int out_size, void* d_ws,
                              size_t ws_size, hipStream_t stream) {
  (void)in_sizes; (void)n_in; (void)out_size; (void)ws_size;
  const float* q = (const float*)d_in[0];
  const float* k = (const float*)d_in[1];
  const float* v = (const float*)d_in[2];
  const float* mask = (const float*)d_in[3];
  const float* wq_w = (const float*)d_in[4];
  const float* wq_b = (const float*)d_in[5];
  const float* dense_w = (const float*)d_in[6];
  const float* dense_b = (const float*)d_in[7];

  float* out = (float*)d_out;                          // [B,S,128]
  float* attn = out + (size_t)BATCH * SEQ * D_MODEL;   // [B,H,S,S]

  const size_t headElems = (size_t)BATCH * NUM_HEADS * SEQ * DEPTH;  // 1M
  float* qh = (float*)d_ws;           // 4 MiB each
  float* kh = qh + headElems;
  float* vh = kh + headElems;
  float* ctx = vh + headElems;        // [B,H,S,16] == buggy-reshape flat rows

  dim3 blk(256);
  proj_to_heads<<<512, blk, 0, stream>>>(q, wq_w, wq_b, qh);
  proj_to_heads<<<512, blk, 0, stream>>>(k, wq_w, wq_b, kh);
  proj_to_heads<<<512, blk, 0, stream>>>(v, wq_w, wq_b, vh);

  const size_t shmem = (size_t)(16 * SEQ + 256 + 8 * 256) * sizeof(float);
  attention<<<BATCH * NUM_HEADS * (SEQ / 16), blk, shmem, stream>>>(
      qh, kh, vh, mask, attn, ctx);

  dense_out<<<512, blk, 0, stream>>>(ctx, dense_w, dense_b, out);
}